// Monolithic_Transformer_86827058856238
// MI455X (gfx1250) — compile-verified
//
#include <hip/hip_runtime.h>
#include <math.h>

// ---------------------------------------------------------------------------
// MI455X (gfx1250) transformer block.
// Math: full fp32 via V_WMMA_F32_16X16X4_F32 (workload is memory-bound:
// ~175 GFLOP vs ~200MB HBM @ 23.3 TB/s, so fp32 tensor ops are free and exact).
// Data movement: GEMM tiles staged by the Tensor Data Mover
// (tensor_load_to_lds, TENSORcnt-gated, double-buffered) with TDM padding
// reproducing the bank-conflict-free LDS strides.
// ---------------------------------------------------------------------------

typedef float v2f __attribute__((ext_vector_type(2)));
typedef float v8f __attribute__((ext_vector_type(8)));
typedef unsigned int v4u __attribute__((ext_vector_type(4)));
typedef int v4i __attribute__((ext_vector_type(4)));
typedef int v8i __attribute__((ext_vector_type(8)));

// 8-arg form: (neg_a, A, neg_b, B, c_mod, C, reuse_a, reuse_b)
#define WMMA_F32(a, b, c) \
    __builtin_amdgcn_wmma_f32_16x16x4_f32(false, (a), false, (b), (short)0, (c), false, false)

// ---------------------------------------------------------------------------
// TDM 2-D tile load: global [tile_d1 rows x tile_d0 elems, row stride
// stride_elems] -> LDS at lds_off, with per-row LDS padding so the LDS row
// stride becomes tile_d0 + pad DWORDs (f32 elements).
//   pad_interval_code: DWORDs between pads = 2<<code
//   pad_amount_code:   pad DWORDs = code+1
// ---------------------------------------------------------------------------
__device__ __forceinline__ void tdm_load_2d(unsigned lds_off, const float* gaddr,
                                            unsigned tile_d0, unsigned tile_d1,
                                            unsigned long long stride_elems,
                                            unsigned pad_interval_code,
                                            unsigned pad_amount_code)
{
    unsigned long long ga = (unsigned long long)(uintptr_t)gaddr;
    v4u g0;
    g0[0] = 1u;                                               // count=1 (valid), user mode
    g0[1] = lds_off;                                          // lds_addr (bytes)
    g0[2] = (unsigned)(ga & 0xFFFFFFFFu);                     // global_addr[31:0]
    g0[3] = (unsigned)((ga >> 32) & 0x01FFFFFFu) | (2u << 30);// global_addr[56:32] | type=2

    const unsigned td0 = 0x40000000u, td1 = 0x40000000u;      // huge tensor dims: tiles in-bounds
    v8i g1;
    g1[0] = (int)((2u << 16)                                  // data_size = 4B
                | (1u << 20)                                  // pad_enable
                | (pad_interval_code << 22)
                | (pad_amount_code << 25));
    g1[1] = (int)((td0 & 0xFFFFu) << 16);                     // tensor_dim0[15:0]
    g1[2] = (int)(((td0 >> 16) & 0xFFFFu) | ((td1 & 0xFFFFu) << 16));
    g1[3] = (int)(((td1 >> 16) & 0xFFFFu) | (tile_d0 << 16)); // tile_dim0
    g1[4] = (int)(tile_d1 & 0xFFFFu);                         // tile_dim1, tile_dim2=0
    g1[5] = (int)(stride_elems & 0xFFFFFFFFull);              // tensor_dim0_stride[31:0]
    g1[6] = (int)((stride_elems >> 32) & 0xFFFFull);          // stride[47:32], dim1_stride=0
    g1[7] = 0;

    v4i gz = {0, 0, 0, 0};
#if __has_include(<hip/amd_detail/amd_gfx1250_TDM.h>)
    v8i gz8 = {0, 0, 0, 0, 0, 0, 0, 0};
    __builtin_amdgcn_tensor_load_to_lds(g0, g1, gz, gz, gz8, 0);
#else
    __builtin_amdgcn_tensor_load_to_lds(g0, g1, gz, gz, 0);
#endif
}

__device__ __forceinline__ unsigned lds_off_u32(const void* p)
{
    // Flat shared-aperture address: addr[31:0] is the LDS byte offset.
    return (unsigned)(uintptr_t)p;
}

// ---------------------------------------------------------------------------
// Generic GEMM: C[M,N] = act( (A[M,K] @ W[K,N] + bias[N]) * scale )
// 256 threads = 8 wave32; block tile 128x64; BK=16; fp32 WMMA.
// A/B tiles DMA'd by TDM into double-buffered padded LDS; wave 0 issues the
// next tile's loads, the workgroup is gated by s_wait_tensorcnt + barrier,
// so DMA(t+1) overlaps compute(t).
// ---------------------------------------------------------------------------
__launch_bounds__(256)
__global__ void gemm_bias_kernel(const float* __restrict__ A,
                                 const float* __restrict__ W,
                                 const float* __restrict__ bias,
                                 float* __restrict__ C,
                                 int M, int N, int K,
                                 int lda, int ldc,
                                 float scale, int relu)
{
    constexpr int BM = 128, BN = 64, BK = 16;
    // LDS row strides via TDM padding: A rows 16+4=20 f32, B rows 64+8=72 f32.
    __shared__ __align__(16) float As[2][BM][20];
    __shared__ __align__(16) float Bs[2][BK][72];

    const int tid   = threadIdx.x;
    const int lane  = tid & 31;
    const int wave  = tid >> 5;
    const int mlane = lane & 15;
    const int half  = lane >> 4;
    const int kb    = half << 1;          // ISA A/B frag: VGPR0=K{kb}, VGPR1=K{kb+1}

    const int m0 = blockIdx.y * BM;
    const int n0 = blockIdx.x * BN;

    v8f acc[4] = {};

    // Prologue: DMA tile 0 into buffer 0 (one TDM issue per tile, wave 0 only;
    // TDM ignores EXEC and signals TENSORcnt once per instruction).
    if (wave == 0) {
        tdm_load_2d(lds_off_u32(&As[0][0][0]), A + (size_t)m0 * lda,
                    16, 128, (unsigned long long)lda, /*16DW*/3, /*4DW*/3);
        tdm_load_2d(lds_off_u32(&Bs[0][0][0]), W + n0,
                    64, 16, (unsigned long long)N,    /*64DW*/5, /*8DW*/7);
    }

    int cur = 0;
    for (int kt = 0; kt < K; kt += BK, cur ^= 1) {
        if (wave == 0) __builtin_amdgcn_s_wait_tensorcnt(0);   // tile `cur` resident
        __syncthreads();
        if (wave == 0 && (kt + BK) < K) {                      // prefetch next tile
            tdm_load_2d(lds_off_u32(&As[cur ^ 1][0][0]),
                        A + (size_t)m0 * lda + kt + BK,
                        16, 128, (unsigned long long)lda, 3, 3);
            tdm_load_2d(lds_off_u32(&Bs[cur ^ 1][0][0]),
                        W + (size_t)(kt + BK) * N + n0,
                        64, 16, (unsigned long long)N, 5, 7);
        }

#pragma unroll
        for (int kk = 0; kk < 4; ++kk) {
            v2f a;
            a.x = As[cur][wave * 16 + mlane][kk * 4 + kb];
            a.y = As[cur][wave * 16 + mlane][kk * 4 + kb + 1];
#pragma unroll
            for (int nt = 0; nt < 4; ++nt) {
                v2f b;
                b.x = Bs[cur][kk * 4 + kb][nt * 16 + mlane];
                b.y = Bs[cur][kk * 4 + kb + 1][nt * 16 + mlane];
                acc[nt] = WMMA_F32(a, b, acc[nt]);
            }
        }
    }

    // C-fragment layout: VGPR r holds row r (lanes 0-15) and row r+8 (lanes 16-31)
#pragma unroll
    for (int nt = 0; nt < 4; ++nt) {
        int col  = n0 + nt * 16 + mlane;
        float bv = bias[col];
#pragma unroll
        for (int r = 0; r < 8; ++r) {
            int row = m0 + wave * 16 + r + half * 8;
            float v = (acc[nt][r] + bv) * scale;
            if (relu) v = fmaxf(v, 0.0f);
            C[(size_t)row * ldc + col] = v;
        }
    }
}

// ---------------------------------------------------------------------------
// Flash attention, diagonal masked, fp32 WMMA.
// q/k/v: [8192, 1024] = [b*n, h*64+d], q pre-scaled by head_dim^-0.5.
// Grid: (n/128, heads, batch). 8 waves per WG; wave owns a 16-row i-tile.
// Online softmax via shfl_xor reductions over the 16-lane halves (matches the
// C-fragment row split); P re-layout C->A through per-wave LDS scratch.
// ---------------------------------------------------------------------------
__launch_bounds__(256)
__global__ void attn_kernel(const float* __restrict__ Q,
                            const float* __restrict__ Km,
                            const float* __restrict__ Vm,
                            float* __restrict__ O, int ldo)
{
    __shared__ __align__(16) float Ks[16][68];      // [j][d], pad for A-style reads
    __shared__ __align__(16) float Vs[16][72];      // [j][d], pad for B-style reads
    __shared__ __align__(16) float Ps[8][16][20];   // per-wave P tile scratch

    const int tid   = threadIdx.x;
    const int lane  = tid & 31;
    const int wave  = tid >> 5;
    const int mlane = lane & 15;
    const int half  = lane >> 4;
    const int kb    = half << 1;

    const int b = blockIdx.z;
    const int h = blockIdx.y;
    const int i0 = blockIdx.x * 128 + wave * 16;              // this wave's i-tile
    const size_t base = ((size_t)b * 1024) * 1024 + h * 64;   // (b,h) slice in [8192,1024]

    // Preload Q A-fragments for the whole i-tile: 16 K-steps over d=64
    v2f qf[16];
    {
        const float* qrow = Q + base + (size_t)(i0 + mlane) * 1024;
#pragma unroll
        for (int dk = 0; dk < 16; ++dk) {
            qf[dk].x = qrow[dk * 4 + kb];
            qf[dk].y = qrow[dk * 4 + kb + 1];
        }
    }

    v8f oacc[4] = {};
    float mrow[8], lrow[8];
#pragma unroll
    for (int r = 0; r < 8; ++r) { mrow[r] = -INFINITY; lrow[r] = 0.0f; }

    for (int j0 = 0; j0 < 1024; j0 += 16) {
        __syncthreads();
        {   // Cooperative stage of K and V 16x64 tiles (one float4 each per thread)
            int jr = tid >> 4;
            int c4 = (tid & 15) << 2;
            float4 kv = *(const float4*)&Km[base + (size_t)(j0 + jr) * 1024 + c4];
            *(float4*)&Ks[jr][c4] = kv;
            float4 vv = *(const float4*)&Vm[base + (size_t)(j0 + jr) * 1024 + c4];
            *(float4*)&Vs[jr][c4] = vv;
        }
        __syncthreads();

        // S(16x16) = Q_i @ K_j^T : 16 chained fp32 WMMAs over d
        v8f s = {};
#pragma unroll
        for (int dk = 0; dk < 16; ++dk) {
            v2f bf;
            bf.x = Ks[mlane][dk * 4 + kb];
            bf.y = Ks[mlane][dk * 4 + kb + 1];
            s = WMMA_F32(qf[dk], bf, s);
        }

        // Diagonal mask + online softmax (per C-frag row; halves hold rows r / r+8)
#pragma unroll
        for (int r = 0; r < 8; ++r) {
            int irow = i0 + r + half * 8;
            int jcol = j0 + mlane;
            float sv = s[r];
            if (irow == jcol) sv = -INFINITY;

            float t = sv;
            t = fmaxf(t, __shfl_xor(t, 1, 32));
            t = fmaxf(t, __shfl_xor(t, 2, 32));
            t = fmaxf(t, __shfl_xor(t, 4, 32));
            t = fmaxf(t, __shfl_xor(t, 8, 32));
            float mnew  = fmaxf(mrow[r], t);
            float alpha = __expf(mrow[r] - mnew);
            float p     = __expf(sv - mnew);

            float ps = p;
            ps += __shfl_xor(ps, 1, 32);
            ps += __shfl_xor(ps, 2, 32);
            ps += __shfl_xor(ps, 4, 32);
            ps += __shfl_xor(ps, 8, 32);

            lrow[r] = lrow[r] * alpha + ps;
            mrow[r] = mnew;
            oacc[0][r] *= alpha; oacc[1][r] *= alpha;
            oacc[2][r] *= alpha; oacc[3][r] *= alpha;

            Ps[wave][r + half * 8][mlane] = p;   // C-layout -> LDS (wave-private, in-order DS)
        }

        // O(16x64) += P(16x16) @ V(16x64): re-read P in A-frag layout
#pragma unroll
        for (int kk = 0; kk < 4; ++kk) {
            v2f a;
            a.x = Ps[wave][mlane][kk * 4 + kb];
            a.y = Ps[wave][mlane][kk * 4 + kb + 1];
#pragma unroll
            for (int nt = 0; nt < 4; ++nt) {
                v2f bf;
                bf.x = Vs[kk * 4 + kb][nt * 16 + mlane];
                bf.y = Vs[kk * 4 + kb + 1][nt * 16 + mlane];
                oacc[nt] = WMMA_F32(a, bf, oacc[nt]);
            }
        }
    }

    // Normalize and write into concat buffer at column offset h*64
#pragma unroll
    for (int r = 0; r < 8; ++r) {
        float inv = 1.0f / lrow[r];
        int irow = i0 + r + half * 8;
        size_t rowoff = ((size_t)b * 1024 + irow) * ldo + h * 64;
#pragma unroll
        for (int nt = 0; nt < 4; ++nt)
            O[rowoff + nt * 16 + mlane] = oacc[nt][r] * inv;
    }
}

// ---------------------------------------------------------------------------
// Host-side launch sequence (graph-capture safe: only kernel launches).
// Workspace (floats): q[8192*1024] k[..] v[..] cat[8192*1088] h[8192*2048]
//   total ~203 MB.
// ---------------------------------------------------------------------------
extern "C" void kernel_launch(void* const* d_in, const int* in_sizes, int n_in,
                              void* d_out, int out_size, void* d_ws, size_t ws_size,
                              hipStream_t stream)
{
    (void)in_sizes; (void)n_in; (void)out_size; (void)ws_size;

    const float* x   = (const float*)d_in[0];
    const float* Wq  = (const float*)d_in[1];
    const float* bq  = (const float*)d_in[2];
    const float* Wk  = (const float*)d_in[3];
    const float* bk  = (const float*)d_in[4];
    const float* Wv  = (const float*)d_in[5];
    const float* bv  = (const float*)d_in[6];
    const float* Win = (const float*)d_in[7];
    const float* bin = (const float*)d_in[8];
    const float* W1  = (const float*)d_in[9];
    const float* b1  = (const float*)d_in[10];
    const float* W2  = (const float*)d_in[11];
    const float* b2  = (const float*)d_in[12];
    float* out = (float*)d_out;

    const int M = 8 * 1024;                 // all batch tokens as one GEMM M
    float* qbuf = (float*)d_ws;
    float* kbuf = qbuf + (size_t)M * 1024;
    float* vbuf = kbuf + (size_t)M * 1024;
    float* cat  = vbuf + (size_t)M * 1024;  // [M, 1088] = [attn(1024) | in_(64)]
    float* hbuf = cat  + (size_t)M * 1088;  // [M, 2048]

    dim3 blk(256);
    const float scaling = 0.125f;           // head_dim^-0.5 = 64^-0.5

    // Q/K/V projections (q pre-scaled)
    gemm_bias_kernel<<<dim3(1024 / 64, M / 128), blk, 0, stream>>>(
        x, Wq, bq, qbuf, M, 1024, 1024, 1024, 1024, scaling, 0);
    gemm_bias_kernel<<<dim3(1024 / 64, M / 128), blk, 0, stream>>>(
        x, Wk, bk, kbuf, M, 1024, 1024, 1024, 1024, 1.0f, 0);
    gemm_bias_kernel<<<dim3(1024 / 64, M / 128), blk, 0, stream>>>(
        x, Wv, bv, vbuf, M, 1024, 1024, 1024, 1024, 1.0f, 0);
    // in_ projection straight into concat columns [1024,1088)
    gemm_bias_kernel<<<dim3(64 / 64, M / 128), blk, 0, stream>>>(
        x, Win, bin, cat + 1024, M, 64, 1024, 1024, 1088, 1.0f, 0);
    // attention -> concat columns [0,1024)
    attn_kernel<<<dim3(8, 16, 8), blk, 0, stream>>>(qbuf, kbuf, vbuf, cat, 1088);
    // MLP
    gemm_bias_kernel<<<dim3(2048 / 64, M / 128), blk, 0, stream>>>(
        cat, W1, b1, hbuf, M, 2048, 1088, 1088, 2048, 1.0f, 1);
    gemm_bias_kernel<<<dim3(1024 / 64, M / 128), blk, 0, stream>>>(
        hbuf, W2, b2, out, M, 1024, 2048, 2048, 1024, 1.0f, 0);
}